// GyroLabeBolmoEncodeBridge_21071109554588
// MI455X (gfx1250) — compile-verified
//
#include <hip/hip_runtime.h>

// ---------------- constants ----------------
#define EPSF        1e-6f
#define TGT_LOGIT  -1.6094379124341003f   // logit(1/6) = ln(0.2)
#define LOG_HALF   -0.6931471805599453f

// ---------------- workspace layout ----------------
struct WS {
    int   ghist[152];      // 0..63 q | 64..67 fam | 68..131 mic | 132..138 shell | 139..145 qw | 146..151 bits
    float partialC[4096];
    int   partialV[4096];
    float shift;
    int   _pad[3];
    int4  chunkAgg[1024];  // (valid, hasB, trail, nb)
    int   chunkR[1024];    // run carry into each chunk
};

struct Agg { int valid, hasB, trail, nb; };

static __device__ __forceinline__ float logit_clipped(float p) {
    p = fminf(fmaxf(p, EPSF), 1.0f - EPSF);
    return logf(p) - log1pf(-p);
}

// --- CDNA5 async global->LDS copy helpers (ASYNCcnt path) ---
// Low 32 bits of a generic pointer to __shared__ == LDS byte offset on CDNA5
// (flat LDS aperture: {SHARED_BASE[63:32], offset[31:0]}, HW truncates to addr[31:0]).
static __device__ __forceinline__ unsigned lds_off(const void* p) {
    return (unsigned)(unsigned long long)p;
}
static __device__ __forceinline__ void async_load_b128(unsigned lds_dst, const void* gptr) {
    asm volatile("global_load_async_to_lds_b128 %0, %1, off"
                 :: "v"(lds_dst), "v"((unsigned long long)gptr) : "memory");
}
static __device__ __forceinline__ void async_load_b32(unsigned lds_dst, const void* gptr) {
    asm volatile("global_load_async_to_lds_b32 %0, %1, off"
                 :: "v"(lds_dst), "v"((unsigned long long)gptr) : "memory");
}
static __device__ __forceinline__ void wait_asynccnt0() {
#if __has_builtin(__builtin_amdgcn_s_wait_asynccnt)
    __builtin_amdgcn_s_wait_asynccnt(0);
#else
    asm volatile("s_wait_asynccnt 0" ::: "memory");
#endif
}
static __device__ __forceinline__ void wait_tensorcnt0() {
#if __has_builtin(__builtin_amdgcn_s_wait_tensorcnt)
    __builtin_amdgcn_s_wait_tensorcnt(0);
#else
    asm volatile("s_wait_tensorcnt 0" ::: "memory");
#endif
}

// --- CDNA5 Tensor Data Mover: 1-row tile load, global -> LDS (TENSORcnt path) ---
// D# per CDNA5 ISA. tensorElems bounds the read so
// OOB elements (the last block's halo) are hardware zero-filled, matching the
// reference's zero-padded _next().
typedef unsigned v4u __attribute__((ext_vector_type(4)));
typedef unsigned v8u __attribute__((ext_vector_type(8)));

static __device__ __forceinline__ void tdm_load_1d(unsigned lds_addr, const void* gptr,
                                                   unsigned tensorElems, unsigned tileElems) {
    const unsigned long long ga = (unsigned long long)gptr;
    v4u g0;
    g0[0] = 1u;                                                // count=1 (valid user D#)
    g0[1] = lds_addr;                                          // lds_addr[31:0] (bytes)
    g0[2] = (unsigned)ga;                                      // global_addr[31:0]
    g0[3] = ((unsigned)(ga >> 32) & 0x1FFFFFFu) | (2u << 30);  // global_addr[56:32] | type=2
    v8u g1;
    g1[0] = 2u << 16;                                          // data_size=2 (4 bytes); mask/flags=0
    g1[1] = (tensorElems & 0xFFFFu) << 16;                     // tensor_dim0[15:0]  (bits 63:48)
    g1[2] = ((tensorElems >> 16) & 0xFFFFu) | (1u << 16);      // tensor_dim0[31:16] | tensor_dim1=1
    g1[3] = (tileElems & 0xFFFFu) << 16;                       // tensor_dim1[31:16]=0 | tile_dim0
    g1[4] = 1u;                                                // tile_dim1=1 | tile_dim2=0
    g1[5] = tileElems;                                         // tensor_dim0_stride[31:0]
    g1[6] = 0u;                                                // stride[47:32] | dim1_stride[15:0]
    g1[7] = 0u;                                                // dim1_stride[47:16]
    asm volatile("tensor_load_to_lds %0, %1" :: "s"(g0), "s"(g1) : "memory");
}

// inclusive Hillis-Steele scan of the segmented run-length monoid over 256 threads
static __device__ __forceinline__ void block_scan_agg(int tid, int* sV, int* sB, int* sT, int* sN) {
    for (int off = 1; off < 256; off <<= 1) {
        int pv = 0, pb = 0, pt = 0, pn = 0;
        const bool act = tid >= off;
        if (act) { pv = sV[tid - off]; pb = sB[tid - off]; pt = sT[tid - off]; pn = sN[tid - off]; }
        __syncthreads();
        if (act) {
            int cv = sV[tid], cb = sB[tid], ct = sT[tid], cn = sN[tid];
            sV[tid] = pv + cv;
            sN[tid] = pn + cn;
            sB[tid] = pb | cb;
            sT[tid] = cb ? ct : (pt + cv);
        }
        __syncthreads();
    }
}

// ---------------- K0: zero global hist accumulators ----------------
__global__ void k0_zero(WS* ws) {
    if (threadIdx.x < 152) ws->ghist[threadIdx.x] = 0;
}

// ---------------- K1: structural + combined + histograms + partial sums ----------------
// st/q tiles arrive via the Tensor Data Mover (TENSORcnt), fam via per-lane async
// copies (ASYNCcnt); mic/valid/blp stream directly through B128 vector loads.
__global__ void k1_structural(const float* __restrict__ blp, const int* __restrict__ st,
                              const int* __restrict__ q,   const int* __restrict__ fam,
                              const int* __restrict__ mic, const int* __restrict__ vmask,
                              float* __restrict__ out, WS* __restrict__ ws, int N) {
    const int  tid       = threadIdx.x;
    const long blockBase = (long)blockIdx.x * 1024;
    const long base      = blockBase + (long)tid * 4;
    const long o_struct  = 2L * (long)N;

    __shared__ int tileSt[1028];
    __shared__ int tileQ [1028];
    __shared__ int tileF [1028];
    __shared__ int sh[152];

    for (int i = tid; i < 152; i += 256) sh[i] = 0;

    // wave 0 issues one TDM descriptor per stream: 1025-element tile (incl. halo),
    // bounded by tensor_dim0 = N - blockBase so the tail halo is zero-filled by HW.
    if (tid == 0) {
        const unsigned rem = (unsigned)(N - blockBase);
        tdm_load_1d(lds_off(tileSt), st + blockBase, rem, 1025u);
        tdm_load_1d(lds_off(tileQ),  q  + blockBase, rem, 1025u);
        wait_tensorcnt0();
    }

    // fam: per-lane async DMA (independent ASYNCcnt queue overlaps the TDM stream)
    async_load_b128(lds_off(&tileF[tid * 4]), fam + base);
    if (tid == 0) {
        if (blockBase + 1024 < N) async_load_b32(lds_off(&tileF[1024]), fam + blockBase + 1024);
        else                      tileF[1024] = 0;
    }

    // overlap: direct vector loads of the per-element-only streams
    __builtin_prefetch(mic + base + 4096, 0, 0);
    int4   m4 = *(const int4*)  (mic   + base);
    int4   v4 = *(const int4*)  (vmask + base);
    float4 b4 = *(const float4*)(blp   + base);

    wait_asynccnt0();
    __syncthreads();

    int sts[5], qs[5], fs[5];
#pragma unroll
    for (int k = 0; k < 5; ++k) {
        sts[k] = tileSt[tid * 4 + k];
        qs[k]  = tileQ [tid * 4 + k];
        fs[k]  = tileF [tid * 4 + k];
    }
    int   ms[4] = { m4.x, m4.y, m4.z, m4.w };
    int   vs[4] = { v4.x, v4.y, v4.z, v4.w };
    float bs[4] = { b4.x, b4.y, b4.z, b4.w };

    int c6[5];
#pragma unroll
    for (int k = 0; k < 5; ++k) { int w = sts[k] & 0xFFF; c6[k] = ((w >> 6) ^ w) & 63; }

    float csum = 0.0f; int vcnt = 0;
    float strA[4], cmbA[4];

#pragma unroll
    for (int k = 0; k < 4; ++k) {
        float d_chi = fminf((float)__popc(c6[k] ^ c6[k + 1]) * (1.0f / 6.0f), 1.0f);
        float d_q   = (float)__popc((qs[k] & 63) ^ (qs[k + 1] & 63)) * (1.0f / 6.0f);
        int   fx    = (fs[k] & 3) ^ (fs[k + 1] & 3);
        float d_fam = 0.5f * ((float)((fx & 1) != 0) + (float)((fx & 2) != 0));
        float score = 0.5f * d_chi + 0.35f * d_q + 0.15f * d_fam;
        score = fminf(fmaxf(score, EPSF), 1.0f);

        const int   vi         = (vs[k] != 0);
        const float structural = vi ? score : 0.0f;
        const float cosine     = expf(fminf(bs[k], 0.0f));
        const float comb       = 0.5f * cosine + 0.5f * structural;   // (WC*cos + WS*str)/(WC+WS)

        strA[k] = structural;
        cmbA[k] = comb;

        if (vi) {
            csum += comb; ++vcnt;
            const int qq = qs[k] & 63;
            atomicAdd(&sh[qq], 1);
            atomicAdd(&sh[64  + (fs[k] & 3)], 1);
            atomicAdd(&sh[68  + (ms[k] & 63)], 1);
            atomicAdd(&sh[132 + __popc(c6[k])], 1);
            atomicAdd(&sh[139 + __popc(qq)], 1);
#pragma unroll
            for (int j = 0; j < 6; ++j)
                if ((qq >> j) & 1) atomicAdd(&sh[146 + j], 1);
        }
    }

    // B128 stores
    *(float4*)(out + o_struct + base) = make_float4(strA[0], strA[1], strA[2], strA[3]);
    *(float4*)(out + base)            = make_float4(cmbA[0], cmbA[1], cmbA[2], cmbA[3]); // stash combined

    // deterministic fixed-order block reduction of partial sums
    __shared__ float rf[256];
    __shared__ int   ri[256];
    rf[tid] = csum; ri[tid] = vcnt;
    __syncthreads();
    for (int s = 128; s > 0; s >>= 1) {
        if (tid < s) { rf[tid] += rf[tid + s]; ri[tid] += ri[tid + s]; }
        __syncthreads();
    }
    if (tid == 0) { ws->partialC[blockIdx.x] = rf[0]; ws->partialV[blockIdx.x] = ri[0]; }

    for (int i = tid; i < 152; i += 256)
        if (sh[i]) atomicAdd(&ws->ghist[i], sh[i]);
}

// ---------------- K2: finalize shift, emit histograms ----------------
__global__ void k2_reduce(WS* __restrict__ ws, float* __restrict__ out, long o_hist, int nPart) {
    const int tid = threadIdx.x;
    __shared__ float rf[256];
    __shared__ int   ri[256];
    float c = 0.0f; int vc = 0;
    for (int i = tid; i < nPart; i += 256) { c += ws->partialC[i]; vc += ws->partialV[i]; }
    rf[tid] = c; ri[tid] = vc;
    __syncthreads();
    for (int s = 128; s > 0; s >>= 1) {
        if (tid < s) { rf[tid] += rf[tid + s]; ri[tid] += ri[tid + s]; }
        __syncthreads();
    }
    if (tid == 0) {
        float denom = fmaxf((float)ri[0], 1.0f);
        float cur   = rf[0] / denom;
        cur = fminf(fmaxf(cur, 1e-4f), 1.0f - 1e-4f);
        ws->shift = TGT_LOGIT - (logf(cur) - log1pf(-cur));
    }
    if (tid < 152) out[o_hist + tid] = (float)ws->ghist[tid];
}

// ---------------- K3a: per-chunk run-length aggregates ----------------
__global__ void k3a_chunkagg(const float* __restrict__ comb, const int* __restrict__ vmask,
                             WS* __restrict__ ws, int N) {
    const int  T    = 16;
    const int  tid  = threadIdx.x;
    const long base = ((long)blockIdx.x * 256 + tid) * T;
    const float shift = ws->shift;

    int valid = 0, nb = 0, hasB = 0, run = 0;
#pragma unroll
    for (int k = 0; k < T; k += 4) {
        float4 c4 = *(const float4*)(comb  + base + k);
        int4   v4 = *(const int4*)  (vmask + base + k);
        float cs[4] = { c4.x, c4.y, c4.z, c4.w };
        int   vs[4] = { v4.x, v4.y, v4.z, v4.w };
#pragma unroll
        for (int j = 0; j < 4; ++j) {
            const int   vi = (vs[j] != 0);
            const float t  = logit_clipped(cs[j]) + shift;     // bmask <=> t >= 0
            if (vi) {
                ++valid; ++run;
                if (t >= 0.0f) { ++nb; hasB = 1; run = 0; }
            }
        }
    }

    __shared__ int sV[256], sB[256], sT[256], sN[256];
    sV[tid] = valid; sB[tid] = hasB; sT[tid] = run; sN[tid] = nb;
    __syncthreads();
    block_scan_agg(tid, sV, sB, sT, sN);
    if (tid == 255) {
        int4 a; a.x = sV[255]; a.y = sB[255]; a.z = sT[255]; a.w = sN[255];
        ws->chunkAgg[blockIdx.x] = a;
    }
}

// ---------------- K3b: scan chunk aggregates -> carries, trailing, patch_count ----------------
__global__ void k3b_carry(WS* __restrict__ ws, float* __restrict__ out, long o_trail, long o_pc) {
    const int tid = threadIdx.x;
    Agg loc[4];
    Agg a; a.valid = 0; a.hasB = 0; a.trail = 0; a.nb = 0;
#pragma unroll
    for (int j = 0; j < 4; ++j) {
        int4 t = ws->chunkAgg[tid * 4 + j];
        Agg b; b.valid = t.x; b.hasB = t.y; b.trail = t.z; b.nb = t.w;
        loc[j] = b;
        Agg r;
        r.valid = a.valid + b.valid; r.nb = a.nb + b.nb; r.hasB = a.hasB | b.hasB;
        r.trail = b.hasB ? b.trail : (a.trail + b.valid);
        a = r;
    }

    __shared__ int sV[256], sB[256], sT[256], sN[256];
    sV[tid] = a.valid; sB[tid] = a.hasB; sT[tid] = a.trail; sN[tid] = a.nb;
    __syncthreads();
    block_scan_agg(tid, sV, sB, sT, sN);

    Agg p; p.valid = 0; p.hasB = 0; p.trail = 0; p.nb = 0;
    if (tid > 0) { p.valid = sV[tid - 1]; p.hasB = sB[tid - 1]; p.trail = sT[tid - 1]; p.nb = sN[tid - 1]; }
#pragma unroll
    for (int j = 0; j < 4; ++j) {
        ws->chunkR[tid * 4 + j] = p.trail;   // run carry into chunk (from global start, trail == carry)
        Agg b = loc[j], r;
        r.valid = p.valid + b.valid; r.nb = p.nb + b.nb; r.hasB = p.hasB | b.hasB;
        r.trail = b.hasB ? b.trail : (p.trail + b.valid);
        p = r;
    }
    if (tid == 255) {
        int trail = sT[255], nb = sN[255];
        out[o_trail] = (float)trail;
        out[o_pc]    = (float)(nb + (trail > 0 ? 1 : 0));
    }
}

// ---------------- K3c: emit logprobs, bmask, lengths ----------------
__global__ void k3c_emit(const int* __restrict__ vmask, float* __restrict__ out,
                         WS* __restrict__ ws, int N) {
    const int  T    = 16;
    const int  tid  = threadIdx.x;
    const long base = ((long)blockIdx.x * 256 + tid) * T;
    const float shift = ws->shift;
    const long o_bm  = (long)N;
    const long o_len = 3L * (long)N + 152;

    __builtin_prefetch(vmask + base + 8192, 0, 0);

    float tv[16]; int vs[16];
#pragma unroll
    for (int k = 0; k < T; k += 4) {
        float4 c4 = *(const float4*)(out   + base + k);   // stashed combined
        int4   v4 = *(const int4*)  (vmask + base + k);
        tv[k + 0] = logit_clipped(c4.x) + shift;
        tv[k + 1] = logit_clipped(c4.y) + shift;
        tv[k + 2] = logit_clipped(c4.z) + shift;
        tv[k + 3] = logit_clipped(c4.w) + shift;
        vs[k + 0] = v4.x; vs[k + 1] = v4.y; vs[k + 2] = v4.z; vs[k + 3] = v4.w;
    }

    // pass 1: thread aggregate
    int valid = 0, nb = 0, hasB = 0, run = 0;
#pragma unroll
    for (int k = 0; k < T; ++k) {
        const int vi = (vs[k] != 0);
        if (vi) { ++valid; ++run; if (tv[k] >= 0.0f) { ++nb; hasB = 1; run = 0; } }
    }

    __shared__ int sV[256], sB[256], sT[256], sN[256];
    sV[tid] = valid; sB[tid] = hasB; sT[tid] = run; sN[tid] = nb;
    __syncthreads();
    block_scan_agg(tid, sV, sB, sT, sN);

    int pv = 0, pb = 0, pt = 0;
    if (tid > 0) { pv = sV[tid - 1]; pb = sB[tid - 1]; pt = sT[tid - 1]; }
    const int chunkR = ws->chunkR[blockIdx.x];
    int r = pb ? pt : (chunkR + pv);   // incoming run for this thread

    // pass 2: emit with B128 stores
#pragma unroll
    for (int k = 0; k < T; k += 4) {
        float clpA[4], bmA[4], lenA[4];
#pragma unroll
        for (int j = 0; j < 4; ++j) {
            const float t   = tv[k + j];
            float sig = 1.0f / (1.0f + expf(-t));
            sig = fminf(fmaxf(sig, EPSF), 1.0f - EPSF);
            const float clp = logf(sig);
            const int   bm  = (t >= 0.0f);          // == (clp >= LOG_HALF)
            const int   vi  = (vs[k + j] != 0);
            if (vi) ++r;
            float len = 0.0f;
            if (bm && vi) { len = (float)r; r = 0; }
            clpA[j] = clp;
            bmA[j]  = bm ? 1.0f : 0.0f;
            lenA[j] = len;
        }
        *(float4*)(out + base + k)         = make_float4(clpA[0], clpA[1], clpA[2], clpA[3]);
        *(float4*)(out + o_bm + base + k)  = make_float4(bmA[0],  bmA[1],  bmA[2],  bmA[3]);
        *(float4*)(out + o_len + base + k) = make_float4(lenA[0], lenA[1], lenA[2], lenA[3]);
    }
}

// ---------------- launch ----------------
extern "C" void kernel_launch(void* const* d_in, const int* in_sizes, int n_in,
                              void* d_out, int out_size, void* d_ws, size_t ws_size,
                              hipStream_t stream) {
    const float* blp  = (const float*)d_in[0];
    const int*   st   = (const int*)d_in[1];
    const int*   q    = (const int*)d_in[2];
    const int*   fam  = (const int*)d_in[3];
    const int*   mic  = (const int*)d_in[4];
    const int*   vm   = (const int*)d_in[5];
    float*       out  = (float*)d_out;
    WS*          ws   = (WS*)d_ws;

    const int N        = in_sizes[0];          // 4 * 1048576 = 4194304
    const int blocksK1 = N / 1024;             // 4096
    const int C        = N / 4096;             // 1024 chunks of 4096 elems

    const long o_hist  = 3L * (long)N;
    const long o_trail = 4L * (long)N + 152;
    const long o_pc    = 4L * (long)N + 153;

    k0_zero      <<<1,        256, 0, stream>>>(ws);
    k1_structural<<<blocksK1, 256, 0, stream>>>(blp, st, q, fam, mic, vm, out, ws, N);
    k2_reduce    <<<1,        256, 0, stream>>>(ws, out, o_hist, blocksK1);
    k3a_chunkagg <<<C,        256, 0, stream>>>(out, vm, ws, N);
    k3b_carry    <<<1,        256, 0, stream>>>(ws, out, o_trail, o_pc);
    k3c_emit     <<<C,        256, 0, stream>>>(vm, out, ws, N);
}